// DistillingLayer_84817014162085
// MI455X (gfx1250) — compile-verified
//
#include <hip/hip_runtime.h>
#include <math.h>

// ---------------------------------------------------------------------------
// DistillingLayer: Conv1d(k=3 shared taps, stride2, pad1) -> ELU -> MaxPool1d
// (k=3, stride2, pad1), x: (16, 4096, 512) fp32 -> out: (16, 1024, 512) fp32.
// Pure bandwidth-bound (160 MB @ 23.3 TB/s ~ 7us; ~0.13 GFLOP) -> no WMMA.
// Use CDNA5 async global->LDS copies to stage overlapping stencil rows once.
// ---------------------------------------------------------------------------

#define AS1 __attribute__((address_space(1)))
#define AS3 __attribute__((address_space(3)))
typedef int v4i __attribute__((ext_vector_type(4)));

constexpr int Bsz = 16;
constexpr int Lin = 4096;
constexpr int Dch = 512;
constexpr int Lpool = 1024;         // final output length
constexpr int TL = 16;              // pooled outputs per block (along L)
constexpr int TD = 128;             // channels per block
constexpr int ROWS = 4 * TL + 3;    // 67 staged input L-positions
constexpr int ROWF = TD;            // floats per staged row (128 -> 512B)

__device__ __forceinline__ float elu1(float v) {
    return v > 0.0f ? v : expm1f(v);
}

__global__ __launch_bounds__(256)
void DistillingLayer_84817014162085_kernel(const float* __restrict__ x,
                                           const float* __restrict__ w,
                                           const float* __restrict__ bias,
                                           float* __restrict__ out) {
    __shared__ float lds[ROWS * ROWF];   // 67 * 128 * 4 = 33.5 KB

    const int lane = threadIdx.x & 31;   // float4 channel column (0..31)
    const int wav  = threadIdx.x >> 5;   // wave id (0..7)

    const int lp0 = blockIdx.x * TL;     // first pooled output of this tile
    const int d0  = blockIdx.y * TD;     // first channel of this tile
    const int b   = blockIdx.z;          // batch

    const int lstart = 4 * lp0 - 3;      // first input L-position needed

    // ---- stage x[b, lstart .. lstart+66, d0 : d0+128] into LDS -------------
    for (int r = wav; r < ROWS; r += 8) {
        const int l = lstart + r;
        float* ldst = &lds[r * ROWF + lane * 4];
        if (l >= 0 && l < Lin) {
            const float* gsrc =
                x + ((size_t)b * Lin + l) * Dch + d0 + lane * 4;
#if __has_builtin(__builtin_amdgcn_global_load_async_to_lds_b128)
            __builtin_amdgcn_global_load_async_to_lds_b128(
                (AS1 v4i*)gsrc, (AS3 v4i*)ldst, 0, 0);
#else
            *(float4*)ldst = *(const float4*)gsrc;
#endif
        } else {
            *(float4*)ldst = make_float4(0.f, 0.f, 0.f, 0.f);  // zero pad l<0
        }
    }
#if __has_builtin(__builtin_amdgcn_global_load_async_to_lds_b128)
#if __has_builtin(__builtin_amdgcn_s_wait_asynccnt)
    __builtin_amdgcn_s_wait_asynccnt(0);
#else
    asm volatile("s_wait_asynccnt 0" ::: "memory");
#endif
#endif
    __syncthreads();

    const float w0 = w[0], w1 = w[1], w2 = w[2], bb = bias[0];

    // ---- compute: each thread -> 2 pooled outputs, one float4 of channels --
#pragma unroll
    for (int i = 0; i < 2; ++i) {
        const int lpl = wav + 8 * i;     // 0..15 within tile
        const int lp  = lp0 + lpl;       // global pooled index
        const int rb  = 4 * lpl;         // first LDS row of 7-row window
        const float* base = &lds[rb * ROWF + lane * 4];

        float4 v0 = *(const float4*)(base + 0 * ROWF);
        float4 v1 = *(const float4*)(base + 1 * ROWF);
        float4 v2 = *(const float4*)(base + 2 * ROWF);
        float4 v3 = *(const float4*)(base + 3 * ROWF);
        float4 v4 = *(const float4*)(base + 4 * ROWF);
        float4 v5 = *(const float4*)(base + 5 * ROWF);
        float4 v6 = *(const float4*)(base + 6 * ROWF);

        // conv taps: y[lc] = w0*x[2lc-1] + w1*x[2lc] + w2*x[2lc+1] + b
        // window rows: y[2lp-1] -> v0..v2, y[2lp] -> v2..v4, y[2lp+1] -> v4..v6
        auto conv = [&](const float4& a, const float4& c, const float4& d) {
            float4 r;
            r.x = fmaf(w0, a.x, fmaf(w1, c.x, fmaf(w2, d.x, bb)));
            r.y = fmaf(w0, a.y, fmaf(w1, c.y, fmaf(w2, d.y, bb)));
            r.z = fmaf(w0, a.z, fmaf(w1, c.z, fmaf(w2, d.z, bb)));
            r.w = fmaf(w0, a.w, fmaf(w1, c.w, fmaf(w2, d.w, bb)));
            return r;
        };
        float4 ym = conv(v0, v1, v2);
        float4 y0 = conv(v2, v3, v4);
        float4 y1 = conv(v4, v5, v6);

        // ELU (alpha = 1)
        y0.x = elu1(y0.x); y0.y = elu1(y0.y); y0.z = elu1(y0.z); y0.w = elu1(y0.w);
        y1.x = elu1(y1.x); y1.y = elu1(y1.y); y1.z = elu1(y1.z); y1.w = elu1(y1.w);

        // MaxPool k=3 stride2 pad1: out[lp] = max(y[2lp-1], y[2lp], y[2lp+1]);
        // y[2lp-1] only exists for lp > 0 (pad is -inf -> excluded);
        // y[2lp+1] = y[2047] at lp=1023 is always in range.
        float4 m;
        m.x = fmaxf(y0.x, y1.x);
        m.y = fmaxf(y0.y, y1.y);
        m.z = fmaxf(y0.z, y1.z);
        m.w = fmaxf(y0.w, y1.w);
        if (lp > 0) {
            ym.x = elu1(ym.x); ym.y = elu1(ym.y);
            ym.z = elu1(ym.z); ym.w = elu1(ym.w);
            m.x = fmaxf(m.x, ym.x);
            m.y = fmaxf(m.y, ym.y);
            m.z = fmaxf(m.z, ym.z);
            m.w = fmaxf(m.w, ym.w);
        }

        float* o = out + ((size_t)b * Lpool + lp) * Dch + d0 + lane * 4;
        *(float4*)o = m;
    }
}

extern "C" void kernel_launch(void* const* d_in, const int* in_sizes, int n_in,
                              void* d_out, int out_size, void* d_ws, size_t ws_size,
                              hipStream_t stream) {
    const float* x = (const float*)d_in[0];   // (16, 4096, 512)
    const float* w = (const float*)d_in[1];   // (3,)
    const float* b = (const float*)d_in[2];   // (1,)
    float* out = (float*)d_out;               // (16, 1024, 512)

    dim3 grid(Lpool / TL, Dch / TD, Bsz);     // (64, 4, 16) blocks
    dim3 block(256);                          // 8 wave32s
    DistillingLayer_84817014162085_kernel<<<grid, block, 0, stream>>>(x, w, b, out);
}